// LogisticModel_1348619731597
// MI455X (gfx1250) — compile-verified
//
#include <hip/hip_runtime.h>

// LogisticModel logp: elementwise + time-shift-by-1. HBM-bound:
//   402 MB traffic / 23.3 TB/s  ->  ~17.3 us floor on MI455X.
// 8 f32 elements per lane (2x global_load_b128), 32-bit SADDR-form addressing,
// NT cache hints on single-use streams, wave32 ds_bpermute for the halo.

typedef float v4f __attribute__((ext_vector_type(4)));

#define GAIN_F      2.0f
#define DECAY_F     0.9f
#define INV_NOISE_F 10.0f
// -(log(0.1) + 0.5*log(2*pi)) = 2.3025850929940457 - 0.9189385332046727
#define NEG_LOG_NORM_F 1.3836465597893730f

#define T_LEN   8192        // power of two; multiple of 256 = elems per wave
#define BLOCK   256         // 8 wave32 per workgroup
#define EPT     8           // elements per thread (2 x float4)

__device__ __forceinline__ void logp4(const float* xe, const float* se,
                                      const float* pv, float* rr)
{
#pragma unroll
    for (int i = 0; i < 4; ++i) {
        // sigmoid(2*s) = 1 / (1 + exp(-2*s)): v_exp_f32 + v_rcp_f32 fast path
        float e    = __expf(-GAIN_F * se[i]);
        float bias = __builtin_amdgcn_rcpf(1.0f + e);
        float t    = __builtin_fmaf(-DECAY_F, pv[i], xe[i]);   // x - 0.9*x_prev
        float z    = (t - bias) * INV_NOISE_F;
        rr[i]      = __builtin_fmaf(-0.5f * z, z, NEG_LOG_NORM_F);
    }
}

__global__ __launch_bounds__(BLOCK)
void logistic_logp_kernel(const float* __restrict__ s,
                          const float* __restrict__ x,
                          float* __restrict__ out)
{
    // 32-bit element offset: max index 33.5M < 2^27, byte offset < 2^31,
    // so the backend can use SGPR-base + 32-bit VGPR-offset global loads.
    const unsigned g  = blockIdx.x * BLOCK + threadIdx.x;
    const unsigned e0 = g * EPT;                    // first of 8 elements

    // s: streamed once, no reuse -> non-temporal 128-bit loads
    v4f sA = __builtin_nontemporal_load(reinterpret_cast<const v4f*>(s + e0));
    v4f sB = __builtin_nontemporal_load(reinterpret_cast<const v4f*>(s + e0 + 4));
    // x: regular-temporal loads (last element reused as neighbor's halo)
    v4f xA = *reinterpret_cast<const v4f*>(x + e0);
    v4f xB = *reinterpret_cast<const v4f*>(x + e0 + 4);

    // x_prev for element 0 of this lane's chunk:
    //   lanes 1..31: previous lane's xB.w via wave32 shuffle (ds_bpermute)
    //   lane 0:      one scalar load of x[e0-1], or 0.0 at a row start.
    // T_LEN % (32*EPT) == 0 => row starts always hit lane 0.
    const int lane = (int)(threadIdx.x & (unsigned)(warpSize - 1));
    float up = __shfl_up(xB.w, 1);
    float prev;
    if (lane == 0) {
        const unsigned t0 = e0 & (T_LEN - 1);
        prev = (t0 == 0) ? 0.0f : x[e0 - 1u];
    } else {
        prev = up;
    }

    const float xeA[4] = { xA.x, xA.y, xA.z, xA.w };
    const float seA[4] = { sA.x, sA.y, sA.z, sA.w };
    const float pvA[4] = { prev, xA.x, xA.y, xA.z };
    const float xeB[4] = { xB.x, xB.y, xB.z, xB.w };
    const float seB[4] = { sB.x, sB.y, sB.z, sB.w };
    const float pvB[4] = { xA.w, xB.x, xB.y, xB.z };

    float rA[4], rB[4];
    logp4(xeA, seA, pvA, rA);
    logp4(xeB, seB, pvB, rB);

    v4f oA, oB;
    oA.x = rA[0]; oA.y = rA[1]; oA.z = rA[2]; oA.w = rA[3];
    oB.x = rB[0]; oB.y = rB[1]; oB.z = rB[2]; oB.w = rB[3];

    // output never re-read -> non-temporal 128-bit stores
    __builtin_nontemporal_store(oA, reinterpret_cast<v4f*>(out + e0));
    __builtin_nontemporal_store(oB, reinterpret_cast<v4f*>(out + e0 + 4));
}

extern "C" void kernel_launch(void* const* d_in, const int* in_sizes, int n_in,
                              void* d_out, int out_size, void* d_ws, size_t ws_size,
                              hipStream_t stream)
{
    const float* s   = (const float*)d_in[0];   // [B, T] f32
    const float* x   = (const float*)d_in[1];   // [B, T] f32
    float*       out = (float*)d_out;           // [B, T] f32

    const long long n    = (long long)in_sizes[0];   // B*T = 33,554,432
    const long long nthr = n / EPT;                  // 4,194,304 threads
    const int       grid = (int)((nthr + BLOCK - 1) / BLOCK);  // 16384 blocks

    hipLaunchKernelGGL(logistic_logp_kernel, dim3(grid), dim3(BLOCK), 0, stream,
                       s, x, out);
}